// SelfAttentionBlock_88794153877713
// MI455X (gfx1250) — compile-verified
//
#include <hip/hip_runtime.h>
#include <hip/hip_bf16.h>

// ---------------------------------------------------------------------------
// Graph attention (TransformerConv-style) on MI455X / gfx1250.
// Pipeline:
//   K1: QKV = X @ W + b          -- f32 WMMA (v_wmma_f32_16x16x4_f32), wave32
//   K2: compat[e,h] = (q.k)/4, segment-max into m[node,h]   (atomic max f32)
//   K3: ex = exp(compat - m[s]), segment-sum into den[node,h]
//   K4: out[s] += v[t] * ex/(den[s]+eps)                     (atomic add f32)
// QKV (76.8MB) + out (25.6MB) are L2-resident (192MB L2), so the 1.2GB of
// per-edge gathers are L2-served; HBM traffic is ~200MB total.
// ---------------------------------------------------------------------------

typedef __attribute__((ext_vector_type(2))) float v2f;
typedef __attribute__((ext_vector_type(8))) float v8f;

#define DIM       64
#define NHEADS    4
#define HDIM      16
#define QKV_COLS  192          // 3 * DIM
#define NTILES_N  (QKV_COLS / 16)   // 12
#define QK_SCALE  0.25f        // HEAD_DIM^-0.5 = 16^-0.5
#define EPS_DEN   1e-16f

// -------------------------- K1: QKV projection -----------------------------
// One wave per 16x16 tile of the 100000x192 output. K=64 -> 16 WMMA steps.
__global__ __launch_bounds__(256) void qkv_wmma_kernel(
    const float* __restrict__ X,     // N x 64 row-major
    const float* __restrict__ W,     // 64 x 192 row-major
    const float* __restrict__ bias,  // 192
    float* __restrict__ QKV,         // N x 192 row-major
    int mtiles) {
  const int wave   = (blockIdx.x * blockDim.x + threadIdx.x) >> 5;
  const int lane   = threadIdx.x & 31;
  const int tm     = wave / NTILES_N;
  const int tn     = wave % NTILES_N;
  if (tm >= mtiles) return;           // wave-uniform: EXEC stays all-ones

  const int m      = lane & 15;       // A row / C col index within tile
  const int khalf  = (lane >> 4) << 1;         // 0 or 2 (A/B K sub-block)
  const int ncol   = tn * 16 + (lane & 15);    // B/C column

  const float* xrow = X + (size_t)(tm * 16 + m) * DIM;

  v8f acc = {0.f, 0.f, 0.f, 0.f, 0.f, 0.f, 0.f, 0.f};
  #pragma unroll
  for (int kk = 0; kk < DIM; kk += 4) {
    v2f a, bf;
    // A: lane holds M=m, VGPR j holds K = kk + khalf + j (contiguous pair -> b64 load)
    a.x  = xrow[kk + khalf + 0];
    a.y  = xrow[kk + khalf + 1];
    // B: lane holds N=ncol, VGPR j holds K = kk + khalf + j
    bf.x = W[(size_t)(kk + khalf + 0) * QKV_COLS + ncol];
    bf.y = W[(size_t)(kk + khalf + 1) * QKV_COLS + ncol];
    acc = __builtin_amdgcn_wmma_f32_16x16x4_f32(
        /*neg_a=*/false, a, /*neg_b=*/false, bf,
        /*c_mod=*/(short)0, acc, /*reuse_a=*/false, /*reuse_b=*/false);
  }

  // C layout: VGPR i -> row (lane<16 ? i : i+8), col = lane&15
  const int rbase = tm * 16 + ((lane >> 4) << 3);
  const float bv  = bias[ncol];
  #pragma unroll
  for (int i = 0; i < 8; ++i) {
    QKV[(size_t)(rbase + i) * QKV_COLS + ncol] = acc[i] + bv;
  }
}

// -------------------------- init helpers ------------------------------------
__global__ void fill_u32_kernel(unsigned int* __restrict__ p, unsigned int v, int n) {
  int i = blockIdx.x * blockDim.x + threadIdx.x;
  if (i < n) p[i] = v;
}

// ---- float atomic max via signed/unsigned integer atomics (init = -inf) ----
__device__ inline void atomicMaxF32(float* addr, float val) {
  val += 0.0f;  // normalize -0.0 -> +0.0 so the sign-split trick is safe
  if (val >= 0.0f) {
    atomicMax((int*)addr, __float_as_int(val));
  } else {
    atomicMin((unsigned int*)addr, __float_as_uint(val));
  }
}

// -------------------------- K2: compat + segment max ------------------------
// One thread per (edge, head).
__global__ __launch_bounds__(256) void compat_max_kernel(
    const float* __restrict__ qkv, const int* __restrict__ ei,
    float* __restrict__ compat, float* __restrict__ mbuf, int E) {
  int idx = blockIdx.x * blockDim.x + threadIdx.x;
  if (idx >= E * NHEADS) return;
  const int e = idx >> 2;
  const int h = idx & 3;
  const int s = ei[e];        // source (query) node
  const int t = ei[E + e];    // target (key/value) node

  const float4* q = (const float4*)(qkv + (size_t)s * QKV_COLS + h * HDIM);          // q block
  const float4* k = (const float4*)(qkv + (size_t)t * QKV_COLS + DIM + h * HDIM);    // k block
  float acc = 0.f;
  #pragma unroll
  for (int i = 0; i < 4; ++i) {
    float4 a = q[i], b = k[i];
    acc += a.x * b.x + a.y * b.y + a.z * b.z + a.w * b.w;
  }
  acc *= QK_SCALE;
  compat[idx] = acc;                         // (E, H) row-major
  atomicMaxF32(&mbuf[s * NHEADS + h], acc);
}

// -------------------------- K3: exp + segment sum ---------------------------
__global__ __launch_bounds__(256) void exp_den_kernel(
    const int* __restrict__ ei, float* __restrict__ compat,
    const float* __restrict__ mbuf, float* __restrict__ den, int E) {
  int idx = blockIdx.x * blockDim.x + threadIdx.x;
  if (idx >= E * NHEADS) return;
  const int e = idx >> 2;
  const int h = idx & 3;
  const int s = ei[e];
  float ex = expf(compat[idx] - mbuf[s * NHEADS + h]);
  compat[idx] = ex;                          // overwrite in place with exp()
  atomicAdd(&den[s * NHEADS + h], ex);
}

// -------------------------- K4: weighted scatter ----------------------------
__global__ __launch_bounds__(256) void scatter_kernel(
    const float* __restrict__ qkv, const int* __restrict__ ei,
    const float* __restrict__ ex, const float* __restrict__ den,
    float* __restrict__ out, int E) {
  int idx = blockIdx.x * blockDim.x + threadIdx.x;
  if (idx >= E * NHEADS) return;
  const int e = idx >> 2;
  const int h = idx & 3;
  const int s = ei[e];
  const int t = ei[E + e];
  const float attn = ex[idx] / (den[s * NHEADS + h] + EPS_DEN);

  const float4* v = (const float4*)(qkv + (size_t)t * QKV_COLS + 2 * DIM + h * HDIM);
  float* o = out + (size_t)s * DIM + h * HDIM;
  #pragma unroll
  for (int i = 0; i < 4; ++i) {
    float4 vv = v[i];
    atomicAdd(&o[4 * i + 0], vv.x * attn);
    atomicAdd(&o[4 * i + 1], vv.y * attn);
    atomicAdd(&o[4 * i + 2], vv.z * attn);
    atomicAdd(&o[4 * i + 3], vv.w * attn);
  }
}

// ---------------------------------------------------------------------------
extern "C" void kernel_launch(void* const* d_in, const int* in_sizes, int n_in,
                              void* d_out, int out_size, void* d_ws, size_t ws_size,
                              hipStream_t stream) {
  const float* x  = (const float*)d_in[0];   // N x 64
  const float* W  = (const float*)d_in[1];   // 64 x 192
  const float* b  = (const float*)d_in[2];   // 192
  const int*   ei = (const int*)d_in[3];     // 2 x E
  float* out      = (float*)d_out;           // N x 64

  const int N = in_sizes[0] / DIM;
  const int E = in_sizes[3] / 2;

  // workspace layout
  float* qkv    = (float*)d_ws;                       // N * 192
  float* compat = qkv + (size_t)N * QKV_COLS;         // E * 4   (then holds exp)
  float* mbuf   = compat + (size_t)E * NHEADS;        // N * 4
  float* den    = mbuf + (size_t)N * NHEADS;          // N * 4

  // per-call init (graph-capture safe)
  hipMemsetAsync(out, 0, (size_t)N * DIM * sizeof(float), stream);
  hipMemsetAsync(den, 0, (size_t)N * NHEADS * sizeof(float), stream);
  {
    int nm = N * NHEADS;
    fill_u32_kernel<<<(nm + 255) / 256, 256, 0, stream>>>(
        (unsigned int*)mbuf, 0xFF800000u /* -inf */, nm);
  }

  // K1: QKV projection via f32 WMMA
  {
    const int mtiles = (N + 15) / 16;           // 6250 (N % 16 == 0)
    const int waves  = mtiles * NTILES_N;       // 75000 waves
    const int blocks = (waves * 32 + 255) / 256;
    qkv_wmma_kernel<<<blocks, 256, 0, stream>>>(x, W, b, qkv, mtiles);
  }

  // K2..K4: edge phase, one thread per (edge, head)
  {
    const long long total = (long long)E * NHEADS;
    const int blocks = (int)((total + 255) / 256);
    compat_max_kernel<<<blocks, 256, 0, stream>>>(qkv, ei, compat, mbuf, E);
    exp_den_kernel<<<blocks, 256, 0, stream>>>(ei, compat, mbuf, den, E);
    scatter_kernel<<<blocks, 256, 0, stream>>>(qkv, ei, compat, den, out, E);
  }
}